// EnhancedDistillationLoss_11527692223274
// MI455X (gfx1250) — compile-verified
//
#include <hip/hip_runtime.h>
#include <math.h>
#include <stdint.h>

// EnhancedDistillationLoss for MI455X (gfx1250, wave32).
// Memory-bound streaming reduction: 621 MB @ 23.3 TB/s ~= 27 us floor.
// Single-pass online softmax per row, async global->LDS double-buffered staging.

#define TEMP_F 2.0f
#define CE_W_F 1.0f
#define KL_W_F 0.5f

#ifndef __has_builtin
#define __has_builtin(x) 0
#endif

typedef int v4i __attribute__((vector_size(16)));  // builtin's pointee type

#if __has_builtin(__builtin_amdgcn_global_load_async_to_lds_b128)
#define HAVE_ASYNC_LDS 1
#else
#define HAVE_ASYNC_LDS 0
#endif

#if __has_builtin(__builtin_amdgcn_s_wait_asynccnt)
#define WAIT_ASYNC(n) __builtin_amdgcn_s_wait_asynccnt(n)
#else
#define WAIT_ASYNC(n) asm volatile("s_wait_asynccnt %0" ::"i"(n) : "memory")
#endif

#define NT 256          // 8 wave32 waves per block
#define TILE4 1536      // float4 per LDS tile (24 KB); x2 buffers = 48 KB LDS
#define NCH (TILE4/NT)  // async chunks issued per thread per tile (= 6)

__global__ __launch_bounds__(NT) void distill_row_kernel(
    const float* __restrict__ logits,   // [B*S, V]
    const float* __restrict__ tlp,      // [S]
    const int*   __restrict__ labels,   // [B*S]
    const int*   __restrict__ amask,    // [B*S]
    float* __restrict__ acc,            // [4]: sum_nll, n_valid, kl_masked, mask_sum
    int S, int V)
{
  const int row = blockIdx.x;
  const int tid = threadIdx.x;
  const float* rowp = logits + (size_t)row * (size_t)V;

  // Online accumulators: m = running max, s1 = sum exp(x-m),
  // s2 = sum exp((x-m)/2) (TEMP==2 -> sqrt of the same exp), sx = sum x.
  float m = -3.0e38f, s1 = 0.0f, s2 = 0.0f, sx = 0.0f;

  auto upd = [&](float x) {
    if (x > m) {                 // rare after warm-up (~log(n) events/thread)
      float r = __expf(m - x);
      s1 *= r; s2 *= sqrtf(r); m = x;
    }
    float e = __expf(x - m);
    s1 += e; s2 += sqrtf(e); sx += x;
  };

  // Row base is odd-offset when V is odd -> scalar head to 16B alignment.
  uintptr_t addr = (uintptr_t)rowp;
  int head = (int)(((16u - (addr & 15u)) & 15u) >> 2);
  if (head > V) head = V;
  int nbody4 = (V - head) >> 2;
  int tail_start = head + (nbody4 << 2);

  for (int i = tid; i < head; i += NT) upd(rowp[i]);
  for (int i = tail_start + tid; i < V; i += NT) upd(rowp[i]);

  const float4* body = (const float4*)(rowp + head);

#if HAVE_ASYNC_LDS
  // Double-buffered async global->LDS staging (ASYNCcnt pipeline).
  __shared__ float4 buf[2][TILE4];
  int ntiles = (nbody4 + TILE4 - 1) / TILE4;

  auto issue = [&](int t, int bb) {
    int t0 = t * TILE4;
#pragma unroll
    for (int k = 0; k < NCH; ++k) {
      int j = t0 + tid + k * NT;
      int jj = (j < nbody4) ? j : 0;  // clamp global addr; LDS slot stays unique
      __builtin_amdgcn_global_load_async_to_lds_b128(
          (v4i*)(body + jj), (v4i*)&buf[bb][tid + k * NT], 0, 0);
    }
  };

  if (ntiles > 0) issue(0, 0);
  for (int t = 0; t < ntiles; ++t) {
    if (t + 1 < ntiles) {
      issue(t + 1, (t + 1) & 1);
      WAIT_ASYNC(NCH);   // async loads complete in order: tile t is done
    } else {
      WAIT_ASYNC(0);
    }
    __syncthreads();
    int cnt = nbody4 - t * TILE4; if (cnt > TILE4) cnt = TILE4;
    const float4* src = buf[t & 1];
    for (int j = tid; j < cnt; j += NT) {
      float4 v = src[j];
      upd(v.x); upd(v.y); upd(v.z); upd(v.w);
    }
    __syncthreads();   // all reads done before this buffer is overwritten
  }
#else
  // Fallback: direct b128 streaming with software prefetch.
  for (int j = tid; j < nbody4; j += NT) {
    __builtin_prefetch(body + j + 4 * NT, 0, 0);
    float4 v = body[j];
    upd(v.x); upd(v.y); upd(v.z); upd(v.w);
  }
#endif

  // ---- wave32 tree reduction of (m, s1, s2, sx) ----
  auto combine = [&](float mo, float s1o, float s2o, float sxo) {
    float M = fmaxf(m, mo);
    float a = __expf(m - M), b = __expf(mo - M);
    s1 = s1 * a + s1o * b;
    s2 = s2 * sqrtf(a) + s2o * sqrtf(b);
    m = M; sx += sxo;
  };

  for (int off = 16; off > 0; off >>= 1)
    combine(__shfl_xor(m, off, 32), __shfl_xor(s1, off, 32),
            __shfl_xor(s2, off, 32), __shfl_xor(sx, off, 32));

  __shared__ float rm[8], rs1[8], rs2[8], rsx[8];
  int wave = tid >> 5, lane = tid & 31;
  if (lane == 0) { rm[wave] = m; rs1[wave] = s1; rs2[wave] = s2; rsx[wave] = sx; }
  __syncthreads();

  if (wave == 0) {
    if (lane < 8) { m = rm[lane]; s1 = rs1[lane]; s2 = rs2[lane]; sx = rsx[lane]; }
    else          { m = -3.0e38f; s1 = 0.0f; s2 = 0.0f; sx = 0.0f; }
    for (int off = 4; off > 0; off >>= 1)
      combine(__shfl_xor(m, off, 32), __shfl_xor(s1, off, 32),
              __shfl_xor(s2, off, 32), __shfl_xor(sx, off, 32));

    if (lane == 0) {
      int s_idx = row % S;
      int lab = labels[row];
      float lse  = m + __logf(s1);                        // logsumexp(x)
      float lseT = m * (1.0f / TEMP_F) + __logf(s2);      // logsumexp(x/T)
      float slp_sum = sx * (1.0f / TEMP_F) - (float)V * lseT;

      float nll = 0.0f, vld = 0.0f;
      if (lab != -100) {
        int ls = lab < 0 ? 0 : lab;
        nll = lse - rowp[ls];
        vld = 1.0f;
      }
      float prob = fminf(__expf(tlp[s_idx]), 0.99f);
      float p = (1.0f - prob) / (float)V;
      float kl_tok = (float)V * p * __logf(p) - p * slp_sum;
      float msk = (float)amask[row];

      atomicAdd(&acc[0], nll);
      atomicAdd(&acc[1], vld);
      atomicAdd(&acc[2], kl_tok * msk);
      atomicAdd(&acc[3], msk);
    }
  }
}

__global__ void distill_init_kernel(float* acc) {
  if (threadIdx.x < 4) acc[threadIdx.x] = 0.0f;
}

__global__ void distill_final_kernel(const float* __restrict__ acc,
                                     float* __restrict__ out) {
  float ce = acc[0] / fmaxf(acc[1], 1.0f);
  float kl = (acc[3] > 0.0f ? acc[2] / acc[3] : acc[2]) * (TEMP_F * TEMP_F);
  out[0] = CE_W_F * ce + KL_W_F * kl;
}

extern "C" void kernel_launch(void* const* d_in, const int* in_sizes, int n_in,
                              void* d_out, int out_size, void* d_ws, size_t ws_size,
                              hipStream_t stream) {
  const float* logits = (const float*)d_in[0];   // [B,S,V] f32
  const float* tlp    = (const float*)d_in[1];   // [S] f32
  const int*   labels = (const int*)d_in[2];     // [B,S] i32
  const int*   amask  = (const int*)d_in[3];     // [B,S] i32
  float* out = (float*)d_out;
  float* acc = (float*)d_ws;

  int S  = in_sizes[1];          // 512
  int BS = in_sizes[2];          // B*S = 1024
  int V  = in_sizes[0] / BS;     // 151643

  distill_init_kernel<<<1, 32, 0, stream>>>(acc);
  distill_row_kernel<<<BS, NT, 0, stream>>>(logits, tlp, labels, amask, acc, S, V);
  distill_final_kernel<<<1, 1, 0, stream>>>(acc, out);
}